// DecoderCell_9586367005129
// MI455X (gfx1250) — compile-verified
//
#include <hip/hip_runtime.h>
#include <hip/hip_bf16.h>
#include <math.h>

// Problem constants
#define Bsz 256
#define Ssz 512
#define Dsz 1024
#define Vsz 32003

typedef float v2f __attribute__((ext_vector_type(2)));
typedef float v4f __attribute__((ext_vector_type(4)));
typedef float v8f __attribute__((ext_vector_type(8)));

// ---------------------------------------------------------------------------
// Generic fp32 WMMA GEMM:  C[M,N] = op( A[M,K] @ W[N,K]^T + bias )
// One wave computes a 16x128 macro-tile (8 N-tiles) with V_WMMA_F32_16X16X4_F32.
// Ping-pong double-buffered K-loop (2 steps per iteration, no register copies):
// while the 8 WMMAs of one buffer execute, the 9 fragments of the other buffer
// are in flight. K/4 must be even (true for K = 1024, 2048).
// A-loader optionally gathers rows through `tokens` (embedding lookup fusion).
// EPI: 0 = raw store, 1 = +bias, 2 = +bias then celu(alpha=1)
// ---------------------------------------------------------------------------
template <int EPI>
__global__ __launch_bounds__(256) void gemm_wmma_f32(
    const float* __restrict__ A, int lda,
    const int* __restrict__ tokens,               // nullptr -> identity rows
    const float* __restrict__ W, int ldw,
    const float* __restrict__ bias,
    float* __restrict__ C, int ldc,
    int M, int N, int K)
{
  constexpr int NTJ = 8;                          // 8 N-tiles (128 cols) per wave
  const int lane   = threadIdx.x & 31;
  const int waveId = threadIdx.x >> 5;
  const int NG     = (N + 127) >> 7;              // 128-wide N groups
  const int totalWaves = (M >> 4) * NG;
  const int wg = blockIdx.x * (blockDim.x >> 5) + waveId;
  if (wg >= totalWaves) return;                   // wave-uniform: EXEC stays all-1 for WMMA

  const int nGroup = wg % NG;
  const int mTile  = wg / NG;                     // consecutive waves share the A tile (L2 reuse)
  const int half   = lane >> 4;                   // lane group (0: lanes 0-15, 1: lanes 16-31)
  const int r      = lane & 15;

  // A fragment source row for this lane (M = mTile*16 + r)
  const int m = mTile * 16 + r;
  const float* Arow = tokens ? (A + (size_t)tokens[m] * lda)
                             : (A + (size_t)m * lda);

  // B fragment source rows: B[k][n] = W[n][k]; column n = n0 + 16*j + r
  const int n0 = nGroup * (16 * NTJ);
  const float* Wrow[NTJ];
  int ncol[NTJ];
#pragma unroll
  for (int j = 0; j < NTJ; ++j) {
    int n = n0 + 16 * j + r;
    ncol[j] = n;
    int nc = (n < N) ? n : (N - 1);               // clamp: garbage column, never stored
    Wrow[j] = W + (size_t)nc * ldw;
  }

  v8f acc[NTJ];
#pragma unroll
  for (int j = 0; j < NTJ; ++j)
#pragma unroll
    for (int i = 0; i < 8; ++i) acc[j][i] = 0.0f;

  // Fragment loader: A frag lanes0-15 hold K=k0,k0+1; lanes16-31 K=k0+2,k0+3.
  // B frag mirrors it across W rows. Each fragment is one b64 load.
  auto ldfrags = [&](int k0, v2f& a, v2f (&b)[NTJ]) {
    a = *(const v2f*)(Arow + k0 + 2 * half);
#pragma unroll
    for (int j = 0; j < NTJ; ++j)
      b[j] = *(const v2f*)(Wrow[j] + k0 + 2 * half);
  };
  auto domma = [&](const v2f& a, const v2f (&b)[NTJ]) {
#pragma unroll
    for (int j = 0; j < NTJ; ++j)
      acc[j] = __builtin_amdgcn_wmma_f32_16x16x4_f32(
          false, a, false, b[j], (short)0, acc[j], false, false);
  };

  // Ping-pong pipeline, two K-steps (8 k-values) per iteration, last pair peeled.
  v2f a0, a1, b0[NTJ], b1[NTJ];
  ldfrags(0, a0, b0);
  int k0 = 0;
  for (; k0 < K - 8; k0 += 8) {
    if ((k0 & 31) == 0 && k0 + 64 < K) {          // gfx1250 prefetch, 256B ahead
      __builtin_prefetch(Arow + k0 + 64, 0, 3);
#pragma unroll
      for (int j = 0; j < NTJ; ++j) __builtin_prefetch(Wrow[j] + k0 + 64, 0, 3);
    }
    ldfrags(k0 + 4, a1, b1);                      // fill buffer 1
    domma(a0, b0);                                // consume buffer 0
    ldfrags(k0 + 8, a0, b0);                      // refill buffer 0
    domma(a1, b1);                                // consume buffer 1
  }
  ldfrags(k0 + 4, a1, b1);                        // final pair
  domma(a0, b0);
  domma(a1, b1);

  // Store: D layout — VGPR i, lane-group half -> row M = i + 8*half, col N = r
#pragma unroll
  for (int j = 0; j < NTJ; ++j) {
    const int n = ncol[j];
    if (n < N) {
      const float bv = (EPI > 0) ? bias[n] : 0.0f;
#pragma unroll
      for (int i = 0; i < 8; ++i) {
        float v = acc[j][i] + bv;
        if (EPI == 2) v = (v > 0.0f) ? v : expm1f(v);   // celu, alpha=1
        C[(size_t)(mTile * 16 + i + 8 * half) * ldc + n] = v;
      }
    }
  }
}

// ---------------------------------------------------------------------------
// GRU activation with h_prev == 0 (first decode step):
//   r = sigmoid(g_r + bih_r + bhh_r); z = sigmoid(g_z + bih_z + bhh_z)
//   n = tanh(g_n + bih_n + r*bhh_n);  h = (1-z)*n
// G: [B, 3D] gate pre-activations (GEMM output, gates ordered r,z,n).
// Vectorized: each thread handles 4 consecutive d (b128 loads/stores).
// ---------------------------------------------------------------------------
__global__ __launch_bounds__(256) void gru_act_kernel(
    const float* __restrict__ G,
    const float* __restrict__ bih, const float* __restrict__ bhh,
    float* __restrict__ H, int ldh)
{
  constexpr int D4 = Dsz / 4;
  const int idx = blockIdx.x * blockDim.x + threadIdx.x;   // over B*D/4
  const int b  = idx / D4;
  const int d4 = idx - b * D4;
  const int d  = d4 * 4;
  const float* g = G + (size_t)b * (3 * Dsz);

  const v4f gr = *(const v4f*)(g + d);
  const v4f gz = *(const v4f*)(g + Dsz + d);
  const v4f gn = *(const v4f*)(g + 2 * Dsz + d);
  const v4f br_i = *(const v4f*)(bih + d);
  const v4f bz_i = *(const v4f*)(bih + Dsz + d);
  const v4f bn_i = *(const v4f*)(bih + 2 * Dsz + d);
  const v4f br_h = *(const v4f*)(bhh + d);
  const v4f bz_h = *(const v4f*)(bhh + Dsz + d);
  const v4f bn_h = *(const v4f*)(bhh + 2 * Dsz + d);

  v4f h;
#pragma unroll
  for (int i = 0; i < 4; ++i) {
    const float ir = gr[i] + br_i[i] + br_h[i];
    const float iz = gz[i] + bz_i[i] + bz_h[i];
    const float rg = 1.0f / (1.0f + expf(-ir));
    const float zg = 1.0f / (1.0f + expf(-iz));
    const float ng = tanhf(gn[i] + bn_i[i] + rg * bn_h[i]);
    h[i] = (1.0f - zg) * ng;
  }
  *(v4f*)(H + (size_t)b * ldh + d) = h;
}

// ---------------------------------------------------------------------------
// Single-pass (online-softmax) attention: one block (8 waves) per batch row.
//   weights[b,s] = h1[b]·enc[b,s];  masked -inf;  summary = softmax(w) @ enc
// enc is streamed exactly once (512 MB total) with b128 loads: each lane holds
// 8 float4s (32 floats) of the 1024-wide row. h1 read from CAT[b, 0:D];
// summary written to CAT[b, D:2D]  (CAT row stride = 2D, forms the concat).
// ---------------------------------------------------------------------------
__global__ __launch_bounds__(256) void attention_kernel(
    const float* __restrict__ enc, const int* __restrict__ encmask,
    float* __restrict__ CAT)
{
  constexpr int NW = 8;                        // waves per block (blockDim 256)
  constexpr int C4 = Dsz / (32 * 4);           // 8 float4s per lane
  const int b    = blockIdx.x;
  const int lane = threadIdx.x & 31;
  const int wave = threadIdx.x >> 5;

  __shared__ v4f  sacc[NW][Dsz / 4];           // 32 KB
  __shared__ float sm[NW], sl[NW];

  // Cache the query row in registers (b128 loads)
  const v4f* h1v = (const v4f*)(CAT + (size_t)b * (2 * Dsz));
  v4f q[C4];
#pragma unroll
  for (int c = 0; c < C4; ++c) q[c] = h1v[lane + 32 * c];

  float m = -INFINITY, l = 0.0f;
  v4f acc[C4];
#pragma unroll
  for (int c = 0; c < C4; ++c)
#pragma unroll
    for (int i = 0; i < 4; ++i) acc[c][i] = 0.0f;

  const float* encb = enc + (size_t)b * Ssz * Dsz;
  for (int s = wave; s < Ssz; s += NW) {
    const v4f* row = (const v4f*)(encb + (size_t)s * Dsz);
    v4f er[C4];
    float dot = 0.0f;
#pragma unroll
    for (int c = 0; c < C4; ++c) {
      er[c] = row[lane + 32 * c];
      dot += er[c][0] * q[c][0] + er[c][1] * q[c][1]
           + er[c][2] * q[c][2] + er[c][3] * q[c][3];
    }
#pragma unroll
    for (int off = 16; off > 0; off >>= 1) dot += __shfl_xor(dot, off, 32);
    if (encmask[b * Ssz + s] == 0) dot = -INFINITY;

    const float mnew = fmaxf(m, dot);
    if (mnew == -INFINITY) continue;           // fully-masked so far
    const float scale = (m == -INFINITY) ? 0.0f : expf(m - mnew);
    const float p     = (dot == -INFINITY) ? 0.0f : expf(dot - mnew);
    l = l * scale + p;
#pragma unroll
    for (int c = 0; c < C4; ++c) acc[c] = acc[c] * scale + er[c] * p;
    m = mnew;
  }

  // Cross-wave combine via LDS
#pragma unroll
  for (int c = 0; c < C4; ++c) sacc[wave][lane + 32 * c] = acc[c];
  if (lane == 0) { sm[wave] = m; sl[wave] = l; }
  __syncthreads();

  float M = -INFINITY;
#pragma unroll
  for (int w = 0; w < NW; ++w) M = fmaxf(M, sm[w]);
  float sc[NW];
  float L = 0.0f;
#pragma unroll
  for (int w = 0; w < NW; ++w) {
    sc[w] = (sm[w] == -INFINITY) ? 0.0f : expf(sm[w] - M);
    L += sc[w] * sl[w];
  }
  const float invL = 1.0f / L;

  // 256 threads x one float4 each = 1024 floats of summary
  const int d4 = threadIdx.x;
  v4f sum;
#pragma unroll
  for (int i = 0; i < 4; ++i) sum[i] = 0.0f;
#pragma unroll
  for (int w = 0; w < NW; ++w) sum += sacc[w][d4] * sc[w];
  *(v4f*)(CAT + (size_t)b * (2 * Dsz) + Dsz + d4 * 4) = sum * invL;
}

// ---------------------------------------------------------------------------
extern "C" void kernel_launch(void* const* d_in, const int* in_sizes, int n_in,
                              void* d_out, int out_size, void* d_ws, size_t ws_size,
                              hipStream_t stream) {
  const int*   tokens   = (const int*)d_in[0];
  const float* enc      = (const float*)d_in[1];
  const int*   encmask  = (const int*)d_in[2];
  const float* dec_emb  = (const float*)d_in[3];
  const float* Wih0     = (const float*)d_in[4];
  // d_in[5] = Whh0: unused (h0_prev == 0 -> gh = bhh0 exactly)
  const float* bih0     = (const float*)d_in[6];
  const float* bhh0     = (const float*)d_in[7];
  const float* Wih1     = (const float*)d_in[8];
  // d_in[9] = Whh1: unused (h1_prev == 0)
  const float* bih1     = (const float*)d_in[10];
  const float* bhh1     = (const float*)d_in[11];
  const float* preout_w = (const float*)d_in[12];
  const float* preout_b = (const float*)d_in[13];
  const float* out_w    = (const float*)d_in[14];
  const float* out_b    = (const float*)d_in[15];
  float* logits = (float*)d_out;

  // Workspace layout (floats): ~7 MB total
  float* G   = (float*)d_ws;                   // [256, 3072] gate pre-activations
  float* H0  = G   + (size_t)Bsz * 3 * Dsz;    // [256, 1024]
  float* CAT = H0  + (size_t)Bsz * Dsz;        // [256, 2048] = concat(h1, summary)
  float* PRE = CAT + (size_t)Bsz * 2 * Dsz;    // [256, 1024]

  auto gemm_blocks = [](int M, int N) {
    int tw = (M / 16) * ((N + 127) / 128);
    return (tw + 7) / 8;                        // 8 waves per 256-thread block
  };

  // 1) GRU0 gates: G = dec_emb[tokens] @ Wih0[:, :D]^T   (prev_att half is zero)
  gemm_wmma_f32<0><<<gemm_blocks(Bsz, 3 * Dsz), 256, 0, stream>>>(
      dec_emb, Dsz, tokens, Wih0, 2 * Dsz, nullptr, G, 3 * Dsz,
      Bsz, 3 * Dsz, Dsz);

  // 2) h0 = gru_act(G, bih0, bhh0)
  gru_act_kernel<<<(Bsz * Dsz / 4) / 256, 256, 0, stream>>>(G, bih0, bhh0, H0, Dsz);

  // 3) GRU1 gates: G = h0 @ Wih1^T
  gemm_wmma_f32<0><<<gemm_blocks(Bsz, 3 * Dsz), 256, 0, stream>>>(
      H0, Dsz, nullptr, Wih1, Dsz, nullptr, G, 3 * Dsz,
      Bsz, 3 * Dsz, Dsz);

  // 4) h1 = gru_act(G, bih1, bhh1), written into CAT[:, 0:D] (stride 2D)
  gru_act_kernel<<<(Bsz * Dsz / 4) / 256, 256, 0, stream>>>(G, bih1, bhh1, CAT, 2 * Dsz);

  // 5) summary = softmax(h1 · enc) @ enc  -> CAT[:, D:2D]   (single pass over enc)
  attention_kernel<<<Bsz, 256, 0, stream>>>(enc, encmask, CAT);

  // 6) PRE = celu(CAT @ preout_w^T + preout_b)
  gemm_wmma_f32<2><<<gemm_blocks(Bsz, Dsz), 256, 0, stream>>>(
      CAT, 2 * Dsz, nullptr, preout_w, 2 * Dsz, preout_b, PRE, Dsz,
      Bsz, Dsz, 2 * Dsz);

  // 7) logits = PRE @ out_w^T + out_b
  gemm_wmma_f32<1><<<gemm_blocks(Bsz, Vsz), 256, 0, stream>>>(
      PRE, Dsz, nullptr, out_w, Dsz, out_b, logits, Vsz,
      Bsz, Vsz, Dsz);
}